// VFE_9122510536945
// MI455X (gfx1250) — compile-verified
//
#include <hip/hip_runtime.h>
#include <hip/hip_bf16.h>

typedef float v2f __attribute__((ext_vector_type(2)));
typedef float v8f __attribute__((ext_vector_type(8)));

#define N_VOX 20000
#define M_PTS 64
#define C_IN  10
#define C0    64
#define C1    128
#define EPS   1e-3f

// workspace float offsets
#define WS_SUMX   0     // 10
#define WS_SXX    16    // 100 (full 10x10)
#define WS_SCALE0 128   // 64
#define WS_BIAS0  192   // 64
#define WS_SUMY   256   // 128
#define WS_SUMQ   384   // 128
#define WS_SCALE1 512   // 128
#define WS_BIAS1  640   // 128
#define WS_TOTAL  768

#define X1_STRIDE 132   // padded LDS row stride (floats) for 64x128 A tile
#define WT_STRIDE 132   // padded LDS row stride for 128x128 W1^T tile

__global__ void vfe_zero(float* __restrict__ ws) {
    int i = blockIdx.x * blockDim.x + threadIdx.x;
    if (i < WS_TOTAL) ws[i] = 0.0f;
}

// Pass over x: accumulate sum_x[10] and sum (x x^T)[10][10] so layer-0 BN stats
// can be derived without materializing y0.
__global__ __launch_bounds__(256) void vfe_stats0(const float* __restrict__ X,
                                                  float* __restrict__ ws, long P) {
    float sx[C_IN] = {};
    float sxx[C_IN * C_IN] = {};
    for (long i = (long)blockIdx.x * blockDim.x + threadIdx.x; i < P;
         i += (long)gridDim.x * blockDim.x) {
        const float* xp = X + i * C_IN;
        float x[C_IN];
#pragma unroll
        for (int c = 0; c < C_IN; ++c) x[c] = xp[c];
#pragma unroll
        for (int c = 0; c < C_IN; ++c) {
            sx[c] += x[c];
#pragma unroll
            for (int d = 0; d < C_IN; ++d) sxx[c * C_IN + d] += x[c] * x[d];
        }
    }
    __shared__ float red[C_IN + C_IN * C_IN];  // 110
    if (threadIdx.x < C_IN + C_IN * C_IN) red[threadIdx.x] = 0.0f;
    __syncthreads();
#pragma unroll
    for (int c = 0; c < C_IN; ++c) atomicAdd(&red[c], sx[c]);
#pragma unroll
    for (int t = 0; t < C_IN * C_IN; ++t) atomicAdd(&red[C_IN + t], sxx[t]);
    __syncthreads();
    int t = threadIdx.x;
    if (t < C_IN)
        atomicAdd(&ws[WS_SUMX + t], red[t]);
    else if (t < C_IN + C_IN * C_IN)
        atomicAdd(&ws[WS_SXX + (t - C_IN)], red[t]);
}

// scale0/bias0 from moments: mean_o = W0_o . mu ; E[y^2]_o = W0_o^T Sigma W0_o
__global__ void vfe_final0(const float* __restrict__ W0, const float* __restrict__ g0,
                           const float* __restrict__ be0, float* __restrict__ ws,
                           float invP) {
    int o = threadIdx.x;
    if (o >= C0) return;
    float w[C_IN];
#pragma unroll
    for (int c = 0; c < C_IN; ++c) w[c] = W0[o * C_IN + c];
    float mean = 0.0f;
#pragma unroll
    for (int c = 0; c < C_IN; ++c) mean += w[c] * ws[WS_SUMX + c];
    mean *= invP;
    float ey2 = 0.0f;
#pragma unroll
    for (int c = 0; c < C_IN; ++c)
#pragma unroll
        for (int d = 0; d < C_IN; ++d) ey2 += w[c] * w[d] * ws[WS_SXX + c * C_IN + d];
    ey2 *= invP;
    float var = ey2 - mean * mean;
    float sc = g0[o] * rsqrtf(var + EPS);
    ws[WS_SCALE0 + o] = sc;
    ws[WS_BIAS0 + o] = be0[o] - mean * sc;
}

__global__ void vfe_final1(const float* __restrict__ g1, const float* __restrict__ be1,
                           float* __restrict__ ws, float invP) {
    int o = threadIdx.x;
    if (o >= C1) return;
    float mean = ws[WS_SUMY + o] * invP;
    float var = ws[WS_SUMQ + o] * invP - mean * mean;
    float sc = g1[o] * rsqrtf(var + EPS);
    ws[WS_SCALE1 + o] = sc;
    ws[WS_BIAS1 + o] = be1[o] - mean * sc;
}

// One voxel per block. Recomputes layer-0 (cheap) into an LDS A-tile
// x1 = [relu(bn(x W0^T)) | repeat(max_M)] (64 x 128), holds W1^T in LDS,
// and runs the layer-1 GEMM with V_WMMA_F32_16X16X4_F32.
// STATS=true : accumulate per-channel sum / sumsq of y1 into ws (for BN1).
// STATS=false: apply BN1+ReLU, max over the 64 rows, write out[n][128].
template <bool STATS>
__global__ __launch_bounds__(256) void vfe_gemm(const float* __restrict__ X,
                                                const float* __restrict__ W0,
                                                const float* __restrict__ W1,
                                                float* __restrict__ ws,
                                                float* __restrict__ out) {
    __shared__ float xr[M_PTS * C_IN];       // 64x10 input rows
    __shared__ float w0s[C0 * C_IN];         // 64x10 layer-0 weights
    __shared__ float s0[C0], b0[C0];
    __shared__ int   m0[C0];                 // layer-0 per-channel max (relu >= 0)
    __shared__ float x1[M_PTS * X1_STRIDE];  // 64x128 A tile (padded)
    __shared__ float wT[C1 * WT_STRIDE];     // W1^T : wT[k][o] = W1[o][k] (padded)
    __shared__ float redS[C1], redQ[C1];     // stats reduction
    __shared__ int   mx[C1];                 // output max reduction
    __shared__ float s1[C1], b1[C1];

    const int tid = threadIdx.x;
    const int n = blockIdx.x;

    for (int i = tid; i < M_PTS * C_IN; i += 256)
        xr[i] = X[(size_t)n * (M_PTS * C_IN) + i];
    for (int i = tid; i < C0 * C_IN; i += 256) w0s[i] = W0[i];
    if (tid < C0) {
        s0[tid] = ws[WS_SCALE0 + tid];
        b0[tid] = ws[WS_BIAS0 + tid];
        m0[tid] = 0;
    }
    if (tid < C1) {
        redS[tid] = 0.0f;
        redQ[tid] = 0.0f;
        mx[tid] = 0;
        if (!STATS) {
            s1[tid] = ws[WS_SCALE1 + tid];
            b1[tid] = ws[WS_BIAS1 + tid];
        }
    }
    // transpose W1 into LDS: wT[k][o] = W1[o*128 + k]
    for (int i = tid; i < C1 * C1; i += 256) {
        int o = i >> 7, k = i & 127;
        wT[k * WT_STRIDE + o] = W1[i];
    }
    __syncthreads();

    // layer-0 pointwise (BN folded into scale/bias) + per-channel max
    for (int i = tid; i < M_PTS * C0; i += 256) {
        int r = i >> 6, o = i & 63;
        float acc = 0.0f;
#pragma unroll
        for (int c = 0; c < C_IN; ++c) acc += xr[r * C_IN + c] * w0s[o * C_IN + c];
        float v = acc * s0[o] + b0[o];
        v = v > 0.0f ? v : 0.0f;
        x1[r * X1_STRIDE + o] = v;
        atomicMax(&m0[o], __float_as_int(v));  // all values >= 0 -> int order ok
    }
    __syncthreads();
    for (int i = tid; i < M_PTS * C0; i += 256) {
        int r = i >> 6, o = i & 63;
        x1[r * X1_STRIDE + C0 + o] = __int_as_float(m0[o]);
    }
    __syncthreads();

    // ---- WMMA GEMM: (64x128) x (128x128) ----
    const int lane = tid & 31;
    const int w = tid >> 5;            // wave 0..7
    const int rt = w & 3;              // row tile (16 rows each)
    const int jbase = (w >> 2) * 4;    // this wave's 4 column tiles
    const int hi = lane >> 4;          // lane half selects K pair
    const int lrow = rt * 16 + (lane & 15);

    // Preload A fragments for all 32 K-steps (reused across 4 column tiles).
    // A 16x4 f32 layout: lane<16 -> K=k0,k0+1 ; lane>=16 -> K=k0+2,k0+3
    v2f aF[32];
#pragma unroll
    for (int k = 0; k < 32; ++k)
        aF[k] = *(const v2f*)&x1[lrow * X1_STRIDE + 4 * k + 2 * hi];

    for (int jj = 0; jj < 4; ++jj) {
        const int j = jbase + jj;
        const int ncol = j * 16 + (lane & 15);
        v8f acc = {};
#pragma unroll
        for (int k = 0; k < 32; ++k) {
            const int kk = 4 * k + 2 * hi;
            v2f b;
            b.x = wT[kk * WT_STRIDE + ncol];
            b.y = wT[(kk + 1) * WT_STRIDE + ncol];
            acc = __builtin_amdgcn_wmma_f32_16x16x4_f32(
                false, aF[k], false, b, (short)0, acc, false, false);
        }
        // C/D layout: lane holds col N=ncol, rows i+8*hi for VGPR i
        if (STATS) {
            float s = 0.0f, q = 0.0f;
#pragma unroll
            for (int i = 0; i < 8; ++i) {
                float v = acc[i];
                s += v;
                q += v * v;
            }
            atomicAdd(&redS[ncol], s);
            atomicAdd(&redQ[ncol], q);
        } else {
            const float sc = s1[ncol], bi = b1[ncol];
            float m = 0.0f;
#pragma unroll
            for (int i = 0; i < 8; ++i) {
                float v = acc[i] * sc + bi;
                v = v > 0.0f ? v : 0.0f;
                m = v > m ? v : m;
            }
            atomicMax(&mx[ncol], __float_as_int(m));
        }
    }
    __syncthreads();

    if (STATS) {
        if (tid < C1) {
            atomicAdd(&ws[WS_SUMY + tid], redS[tid]);
            atomicAdd(&ws[WS_SUMQ + tid], redQ[tid]);
        }
    } else {
        if (tid < C1) out[(size_t)n * C1 + tid] = __int_as_float(mx[tid]);
    }
}

extern "C" void kernel_launch(void* const* d_in, const int* in_sizes, int n_in,
                              void* d_out, int out_size, void* d_ws, size_t ws_size,
                              hipStream_t stream) {
    (void)in_sizes; (void)n_in; (void)out_size; (void)ws_size;
    const float* X   = (const float*)d_in[0];
    const float* W0  = (const float*)d_in[1];
    const float* g0  = (const float*)d_in[2];
    const float* be0 = (const float*)d_in[3];
    const float* W1  = (const float*)d_in[4];
    const float* g1  = (const float*)d_in[5];
    const float* be1 = (const float*)d_in[6];
    float* out = (float*)d_out;
    float* ws  = (float*)d_ws;

    const long P = (long)N_VOX * M_PTS;
    const float invP = 1.0f / (float)P;

    vfe_zero<<<(WS_TOTAL + 255) / 256, 256, 0, stream>>>(ws);
    vfe_stats0<<<1024, 256, 0, stream>>>(X, ws, P);
    vfe_final0<<<1, 64, 0, stream>>>(W0, g0, be0, ws, invP);
    vfe_gemm<true><<<N_VOX, 256, 0, stream>>>(X, W0, W1, ws, out);
    vfe_final1<<<1, 128, 0, stream>>>(g1, be1, ws, invP);
    vfe_gemm<false><<<N_VOX, 256, 0, stream>>>(X, W0, W1, ws, out);
}